// OfficialMamba3SISO_9380208574507
// MI455X (gfx1250) — compile-verified
//
#include <hip/hip_runtime.h>
#include <math.h>

typedef float v2f __attribute__((ext_vector_type(2)));
typedef float v8f __attribute__((ext_vector_type(8)));
typedef unsigned int u32x4 __attribute__((ext_vector_type(4)));
typedef int i32x4 __attribute__((ext_vector_type(4)));
typedef int i32x8 __attribute__((ext_vector_type(8)));

#define BATCH   2
#define SEQLEN  2048
#define DMODEL  1024
#define DINNER  2048
#define NH      32
#define HD      64
#define DS      128
#define NANG    32
#define CHUNK   64
#define NCHUNK  (SEQLEN / CHUNK)
#define DPROJ   4480
#define OFF_Z    0
#define OFF_X    2048
#define OFF_B    4096
#define OFF_C    4224
#define OFF_DT   4352
#define OFF_A    4384
#define OFF_TRAP 4416
#define OFF_ANG  4448

// ---- CDNA5 WMMA f32 16x16x4: D = A(16x4) * B(4x16) + C ----
// A frag: lane(0-15) holds row M=lane, K={0,1}; lane(16-31) row M=lane-16, K={2,3}
// B frag: lane(0-15) holds col N=lane, K={0,1}; lane(16-31) col N=lane-16, K={2,3}
// C frag: vgpr r: lanes 0-15 -> (M=r, N=lane); lanes 16-31 -> (M=r+8, N=lane-16)
__device__ __forceinline__ v8f wmma4(v2f a, v2f b, v8f c) {
  return __builtin_amdgcn_wmma_f32_16x16x4_f32(false, a, false, b, (short)0, c,
                                               false, false);
}

__device__ __forceinline__ float softplus_f(float x) {
  return (x > 20.f) ? x : log1pf(expf(x));
}
__device__ __forceinline__ float sigmoid_f(float x) {
  return 1.f / (1.f + expf(-x));
}

// ------------------------------------------------------------------
// Blocked WMMA GEMM: C(MxN) = A(MxK) @ B(KxN), row-major.
// Block = 8 waves -> 128x128 macro-tile. A staged in LDS (pad 34:
// gcd(34,64)=2, 2*(17*i mod 32) spans 32 distinct even banks -> no
// conflicts on b64 column reads). Each wave: 1 N-tile x 8 M-tiles,
// one B fragment shared by 8 WMMAs per K-step.
// grid.x = N/128, grid.y = M/128
// ------------------------------------------------------------------
__global__ __launch_bounds__(256) void wmma_gemm_f32(const float* __restrict__ A,
                                                     const float* __restrict__ B,
                                                     float* __restrict__ C,
                                                     int M, int N, int K) {
  __shared__ float As[128][34];
  const int lane = threadIdx.x & 31;
  const int wave = threadIdx.x >> 5;
  const int half = lane >> 4;
  const int lrow = lane & 15;
  const int tn = blockIdx.x * 8 + wave;
  const int rowBase = blockIdx.y * 128;
  const float* Bcol = B + (size_t)(2 * half) * N + tn * 16 + lrow;
  v8f acc[8] = {};
  for (int k0 = 0; k0 < K; k0 += 32) {
    __syncthreads();
    for (int idx = threadIdx.x; idx < 128 * 32; idx += 256) {
      const int r = idx >> 5, kk = idx & 31;
      As[r][kk] = A[(size_t)(rowBase + r) * K + k0 + kk];
    }
    __builtin_prefetch(Bcol + (size_t)(k0 + 32) * N, 0, 1);
    __syncthreads();
    for (int kk = 0; kk < 32; kk += 4) {
      v2f bf;
      const size_t boff = (size_t)(k0 + kk) * N;
      bf.x = Bcol[boff];
      bf.y = Bcol[boff + N];
#pragma unroll
      for (int ti = 0; ti < 8; ++ti) {
        v2f a;
        a.x = As[ti * 16 + lrow][kk + 2 * half];
        a.y = As[ti * 16 + lrow][kk + 2 * half + 1];
        acc[ti] = wmma4(a, bf, acc[ti]);
      }
    }
  }
#pragma unroll
  for (int ti = 0; ti < 8; ++ti) {
    float* Cb = C + (size_t)(rowBase + ti * 16 + 8 * half) * N + tn * 16 + lrow;
#pragma unroll
    for (int r = 0; r < 8; ++r) Cb[(size_t)r * N] = acc[ti][r];
  }
}

// ------------------------------------------------------------------
// RMSNorm of Bg/Cg slices of proj, in place. grid = B*L, block = 128
// ------------------------------------------------------------------
__global__ __launch_bounds__(128) void rmsnorm_kernel(const float* __restrict__ Bw,
                                                      const float* __restrict__ Cw,
                                                      float* __restrict__ proj) {
  __shared__ float red[128];
  const size_t row = blockIdx.x;
  const int t = threadIdx.x;
  float* pB = proj + row * DPROJ + OFF_B;
  float* pC = proj + row * DPROJ + OFF_C;
  const float vb = pB[t];
  const float vc = pC[t];
  red[t] = vb * vb;
  __syncthreads();
  for (int s = 64; s > 0; s >>= 1) {
    if (t < s) red[t] += red[t + s];
    __syncthreads();
  }
  const float varb = red[0] * (1.f / DS);
  __syncthreads();
  red[t] = vc * vc;
  __syncthreads();
  for (int s = 64; s > 0; s >>= 1) {
    if (t < s) red[t] += red[t + s];
    __syncthreads();
  }
  const float varc = red[0] * (1.f / DS);
  pB[t] = vb * rsqrtf(varb + 1e-5f) * Bw[t];
  pC[t] = vc * rsqrtf(varc + 1e-5f) * Cw[t];
}

// ------------------------------------------------------------------
// Scalar prep: dt, g1, g2, within-chunk cumsum ell. 1 thread / (b,c,h)
// ------------------------------------------------------------------
__global__ __launch_bounds__(256) void prep_kernel(const float* __restrict__ proj,
                                                   const float* __restrict__ dt_bias,
                                                   float* __restrict__ dtg,
                                                   float* __restrict__ g1g,
                                                   float* __restrict__ g2g,
                                                   float* __restrict__ ellg) {
  const int t = blockIdx.x * 256 + threadIdx.x;
  if (t >= BATCH * NCHUNK * NH) return;
  const int h = t & 31;
  const int c = (t >> 5) & 31;
  const int b = t >> 10;
  const float dtb = dt_bias[h];
  float ell = 0.f;
  for (int j = 0; j < CHUNK; ++j) {
    const int l = c * CHUNK + j;
    const size_t row = (size_t)b * SEQLEN + l;
    const float* pr = proj + row * DPROJ;
    const float dt   = softplus_f(pr[OFF_DT + h] + dtb);
    const float a    = fminf(-softplus_f(pr[OFF_A + h]), -1e-4f);
    const float adt  = a * dt;
    const float beta = sigmoid_f(pr[OFF_TRAP + h]);
    float dtp = 0.f;
    if (l > 0) dtp = softplus_f(proj[(row - 1) * DPROJ + OFF_DT + h] + dtb);
    ell += adt;
    const size_t idx = row * NH + h;
    dtg[idx]  = dt;
    g1g[idx]  = beta * dt;
    g2g[idx]  = (1.f - beta) * expf(adt) * dtp;
    ellg[idx] = ell;
  }
}

// ------------------------------------------------------------------
// Theta cumsum, pass 1: per-chunk partial sums. 1 thread / (b,c,h,i)
// ------------------------------------------------------------------
__global__ __launch_bounds__(256) void theta_partial_kernel(const float* __restrict__ proj,
                                                            const float* __restrict__ dtg,
                                                            float* __restrict__ thsum) {
  const int t = blockIdx.x * 256 + threadIdx.x;
  if (t >= BATCH * NCHUNK * NH * NANG) return;
  const int i = t & 31;
  const int h = (t >> 5) & 31;
  const int c = (t >> 10) & 31;
  const int b = t >> 15;
  float s = 0.f;
  for (int j = 0; j < CHUNK; ++j) {
    const size_t row = (size_t)b * SEQLEN + c * CHUNK + j;
    s += proj[row * DPROJ + OFF_ANG + i] * dtg[row * NH + h];
  }
  thsum[t] = s;
}

// ------------------------------------------------------------------
// Theta cumsum, pass 2: exclusive scan of chunk sums. 1 thr / (b,h,i)
// ------------------------------------------------------------------
__global__ __launch_bounds__(256) void theta_base_kernel(const float* __restrict__ thsum,
                                                         float* __restrict__ thbase) {
  const int t = blockIdx.x * 256 + threadIdx.x;
  if (t >= BATCH * NH * NANG) return;
  const int i = t & 31;
  const int h = (t >> 5) & 31;
  const int b = t >> 10;
  float acc = 0.f;
  for (int c = 0; c < NCHUNK; ++c) {
    const size_t idx = (((size_t)(b * NCHUNK + c) * NH + h) << 5) + i;
    thbase[idx] = acc;
    acc += thsum[idx];
  }
}

// ------------------------------------------------------------------
// RoPE, pass 3: one block(32 lanes) per (b,c,h), 64-step local scan.
// ------------------------------------------------------------------
__global__ __launch_bounds__(32) void rope_kernel(const float* __restrict__ proj,
                                                  const float* __restrict__ dtg,
                                                  const float* __restrict__ thbase,
                                                  const float* __restrict__ Bb,
                                                  const float* __restrict__ Cb,
                                                  float* __restrict__ Kbuf,
                                                  float* __restrict__ Qbuf) {
  const int bid = blockIdx.x;      // (b*NCHUNK + c)*NH + h
  const int h = bid & 31;
  const int c = (bid >> 5) & 31;
  const int b = bid >> 10;
  const int i = threadIdx.x;       // angle index 0..31
  float th = thbase[(((size_t)bid) << 5) + i];
  for (int j = 0; j < CHUNK; ++j) {
    const size_t row = (size_t)b * SEQLEN + c * CHUNK + j;
    const float* pr = proj + row * DPROJ;
    th += pr[OFF_ANG + i] * dtg[row * NH + h];
    const float cth = cosf(th);
    const float sth = sinf(th);
    const float b0 = pr[OFF_B + i]      + Bb[h * DS + i];
    const float b1 = pr[OFF_B + 32 + i] + Bb[h * DS + 32 + i];
    const float b2 = pr[OFF_B + 64 + i] + Bb[h * DS + 64 + i];
    const float b3 = pr[OFF_B + 96 + i] + Bb[h * DS + 96 + i];
    float* Ko = Kbuf + (row * NH + h) * DS;
    Ko[i]      = b0 * cth - b1 * sth;
    Ko[32 + i] = b0 * sth + b1 * cth;
    Ko[64 + i] = b2;
    Ko[96 + i] = b3;
    const float c0 = pr[OFF_C + i]      + Cb[h * DS + i];
    const float c1 = pr[OFF_C + 32 + i] + Cb[h * DS + 32 + i];
    const float c2 = pr[OFF_C + 64 + i] + Cb[h * DS + 64 + i];
    const float c3 = pr[OFF_C + 96 + i] + Cb[h * DS + 96 + i];
    float* Qo = Qbuf + (row * NH + h) * DS;
    Qo[i]      = c0 * cth - c1 * sth;
    Qo[32 + i] = c0 * sth + c1 * cth;
    Qo[64 + i] = c2;
    Qo[96 + i] = c3;
  }
}

// x element of the shift-extended chunk: cidx in [0,64] maps to l = c*64+cidx-1
__device__ __forceinline__ float xval(const float* __restrict__ proj, int b, int c,
                                      int h, int cidx, int p) {
  if (cidx > 64) return 0.f;
  const int ll = c * CHUNK + cidx - 1;
  if (ll < 0) return 0.f;
  return proj[((size_t)b * SEQLEN + ll) * DPROJ + OFF_X + h * HD + p];
}

// ------------------------------------------------------------------
// Chunk kernel: one block per (b,c,h). K_ext tile DMA'd into LDS by
// the Tensor Data Mover (2D strided tile, row stride = NH*DS elems).
//  1. T[i][cidx] = (Q_i . K_ext[cidx]) * cw(i,cidx)   (merged S1/S2)
//  2. y_intra = T @ x_ext
//  3. Wsum = (x_ext * wc)^T @ K_ext ; chunk_decay
// ------------------------------------------------------------------
__global__ __launch_bounds__(256) void chunk_kernel(const float* __restrict__ proj,
                                                    const float* __restrict__ Kbuf,
                                                    const float* __restrict__ Qbuf,
                                                    const float* __restrict__ g1g,
                                                    const float* __restrict__ g2g,
                                                    const float* __restrict__ ellg,
                                                    float* __restrict__ ybuf,
                                                    float* __restrict__ Wsum,
                                                    float* __restrict__ cdk) {
  __shared__ float Ke[65][DS];     // K_ext rows cidx=0..64 (row cidx -> l=c*64+cidx-1)
  __shared__ float T[64][80];      // merged scores, K-dim padded to 80
  __shared__ float ell_s[64], g1_s[64], g2_s[64], wc_s[80];
  const int bid = blockIdx.x;
  const int h = bid & 31;
  const int c = (bid >> 5) & 31;
  const int b = bid >> 10;
  const int tid = threadIdx.x;
  const int lane = tid & 31, wave = tid >> 5;
  const int half = lane >> 4, lrow = lane & 15;

  // --- TDM: DMA K_ext tile (nrows x 128 f32, row stride NH*DS) into LDS ---
  const int nrows = (c == 0) ? 64 : 65;       // c==0: row 0 is the t=-1 zero pad
  if (c == 0) {
    for (int n = tid; n < DS; n += 256) Ke[0][n] = 0.f;
  }
  if (wave == 0) {
    const int startRow = (c == 0) ? 0 : -1;
    const size_t gelem = ((size_t)(b * SEQLEN + c * CHUNK + startRow) * NH + h) * DS;
    const unsigned long long gaddr = (unsigned long long)(const void*)(Kbuf + gelem);
    const unsigned ldsaddr = (unsigned)(size_t)&Ke[(c == 0) ? 1 : 0][0];
    // D# group0: count=1 | lds_addr | global_addr(57b) | type=2
    u32x4 g0;
    g0[0] = 1u;
    g0[1] = ldsaddr;
    g0[2] = (unsigned)gaddr;
    g0[3] = (unsigned)(gaddr >> 32) | (2u << 30);
    // D# group1: data_size=4B(code2); tensor_dim0=128; tensor_dim1=2^20;
    // tile_dim0=128; tile_dim1=nrows; tensor_dim0_stride=NH*DS
    i32x8 g1;
    g1[0] = (int)(2u << 16);
    g1[1] = (int)(((unsigned)DS) << 16);                 // tensor_dim0[15:0]
    g1[2] = (int)(0u);                                   // dim0 hi16=0, dim1 lo16=0
    g1[3] = (int)(0x10u | (((unsigned)DS) << 16));       // dim1 hi16=0x10, tile_dim0
    g1[4] = nrows;                                       // tile_dim1
    g1[5] = NH * DS;                                     // dim0 stride (elements)
    g1[6] = 0;
    g1[7] = 0;
    const i32x4 z4 = {0, 0, 0, 0};
    const i32x8 z8 = {0, 0, 0, 0, 0, 0, 0, 0};
    __builtin_amdgcn_tensor_load_to_lds(g0, g1, z4, z4, z8, 0);
    __builtin_amdgcn_s_wait_tensorcnt(0);
  }
  if (tid < 64) {
    const size_t idx = ((size_t)b * SEQLEN + c * CHUNK + tid) * NH + h;
    ell_s[tid] = ellg[idx];
    g1_s[tid]  = g1g[idx];
    g2_s[tid]  = g2g[idx];
  }
  __syncthreads();
  if (tid < 80) {
    const float ellL = ell_s[63];
    const int cidx = tid, m = cidx - 1;
    float w = 0.f;
    if (m >= 0 && m <= 63) w += g1_s[m] * expf(ellL - ell_s[m]);
    if (cidx <= 63)        w += g2_s[cidx] * expf(ellL - ell_s[cidx]);
    wc_s[cidx] = w;
  }
  if (tid == 0) cdk[bid] = expf(ell_s[63]);
  __syncthreads();

  // ---- phase 1: T = (Q @ Ke^T) * cw,  64x80 output (20 tiles) ----
  const float* Qb = Qbuf + ((size_t)(b * SEQLEN + c * CHUNK) * NH + h) * DS;
  for (int t = wave; t < 20; t += 8) {
    const int ti = t / 5, tj = t % 5;
    int rB = tj * 16 + lrow; if (rB > 64) rB = 64;
    const float* Ar = Qb + (size_t)(ti * 16 + lrow) * (NH * DS) + 2 * half;
    const float* Br = &Ke[rB][2 * half];
    v8f acc = {};
#pragma unroll 4
    for (int k0 = 0; k0 < DS; k0 += 4) {
      v2f a; a.x = Ar[k0]; a.y = Ar[k0 + 1];
      v2f bb; bb.x = Br[k0]; bb.y = Br[k0 + 1];
      acc = wmma4(a, bb, acc);
    }
    const int cidx = tj * 16 + lrow;
#pragma unroll
    for (int r = 0; r < 8; ++r) {
      const int i = ti * 16 + r + 8 * half;
      float cw = 0.f;
      if (cidx <= 64) {
        const int m = cidx - 1;
        if (m >= 0 && i >= m)        cw += g1_s[m] * expf(ell_s[i] - ell_s[m]);
        if (cidx <= 63 && i >= cidx) cw += g2_s[cidx] * expf(ell_s[i] - ell_s[cidx]);
      }
      T[i][cidx] = acc[r] * cw;
    }
  }
  __syncthreads();

  // ---- phase 2: y_intra = T(64x80) @ x_ext(80x64) -> 16 tiles ----
  for (int t = wave; t < 16; t += 8) {
    const int ti = t >> 2, tp = t & 3;
    v8f acc = {};
    for (int k0 = 0; k0 < 80; k0 += 4) {
      const int kA = k0 + 2 * half;
      v2f a; a.x = T[ti * 16 + lrow][kA]; a.y = T[ti * 16 + lrow][kA + 1];
      v2f bb;
      bb.x = xval(proj, b, c, h, kA,     tp * 16 + lrow);
      bb.y = xval(proj, b, c, h, kA + 1, tp * 16 + lrow);
      acc = wmma4(a, bb, acc);
    }
#pragma unroll
    for (int r = 0; r < 8; ++r) {
      const int i = ti * 16 + r + 8 * half;
      const int p = tp * 16 + lrow;
      ybuf[((size_t)(b * SEQLEN + c * CHUNK + i) * NH + h) * HD + p] = acc[r];
    }
  }

  // ---- phase 3: Wsum(64x128) = (x_ext*wc)^T @ Ke -> 32 tiles ----
  for (int t = wave; t < 32; t += 8) {
    const int tp = t >> 3, tn = t & 7;
    v8f acc = {};
    for (int k0 = 0; k0 < 80; k0 += 4) {
      const int kA = k0 + 2 * half;
      v2f a;
      a.x = xval(proj, b, c, h, kA,     tp * 16 + lrow) * wc_s[kA];
      a.y = xval(proj, b, c, h, kA + 1, tp * 16 + lrow) * wc_s[kA + 1];
      int r0 = kA;     if (r0 > 64) r0 = 64;
      int r1 = kA + 1; if (r1 > 64) r1 = 64;
      v2f bb; bb.x = Ke[r0][tn * 16 + lrow]; bb.y = Ke[r1][tn * 16 + lrow];
      acc = wmma4(a, bb, acc);
    }
#pragma unroll
    for (int r = 0; r < 8; ++r) {
      const int p = tp * 16 + r + 8 * half;
      const int n = tn * 16 + lrow;
      Wsum[((size_t)bid * HD + p) * DS + n] = acc[r];
    }
  }
}

// ------------------------------------------------------------------
// Inter-chunk scan + y_inter. One block per (b,h), H(64x128) in LDS.
// ------------------------------------------------------------------
__global__ __launch_bounds__(256) void scan_kernel(const float* __restrict__ Qbuf,
                                                   const float* __restrict__ ellg,
                                                   const float* __restrict__ Wsum,
                                                   const float* __restrict__ cdk,
                                                   float* __restrict__ ybuf) {
  __shared__ float H[64][DS];  // [p][n]
  const int bid = blockIdx.x;
  const int h = bid & 31;
  const int b = bid >> 5;
  const int tid = threadIdx.x;
  float* Hf = &H[0][0];
  for (int idx = tid; idx < 64 * DS; idx += 256) Hf[idx] = 0.f;
  __syncthreads();
  const int lane = tid & 31, wave = tid >> 5;
  const int half = lane >> 4, lrow = lane & 15;
  for (int c = 0; c < NCHUNK; ++c) {
    // y_inter[i][p] = sum_n Q[i,n]*exp(ell_i)*H[p][n]
    for (int t = wave; t < 16; t += 8) {
      const int ti = t >> 2, tp = t & 3;
      const size_t irow = (size_t)(b * SEQLEN + c * CHUNK + ti * 16 + lrow);
      const float ei = expf(ellg[irow * NH + h]);
      const float* Ar = Qbuf + (irow * NH + h) * DS + 2 * half;
      v8f acc = {};
#pragma unroll 4
      for (int k0 = 0; k0 < DS; k0 += 4) {
        v2f a; a.x = Ar[k0] * ei; a.y = Ar[k0 + 1] * ei;
        v2f bb;
        bb.x = H[tp * 16 + lrow][k0 + 2 * half];
        bb.y = H[tp * 16 + lrow][k0 + 2 * half + 1];
        acc = wmma4(a, bb, acc);
      }
#pragma unroll
      for (int r = 0; r < 8; ++r) {
        const int i = ti * 16 + r + 8 * half;
        const int p = tp * 16 + lrow;
        const size_t yi = ((size_t)(b * SEQLEN + c * CHUNK + i) * NH + h) * HD + p;
        ybuf[yi] += acc[r];
      }
    }
    __syncthreads();
    const float cd = cdk[(size_t)((b * NCHUNK + c) * NH) + h];
    const float* Wc = Wsum + (size_t)((b * NCHUNK + c) * NH + h) * (HD * DS);
    for (int idx = tid; idx < 64 * DS; idx += 256)
      Hf[idx] = cd * Hf[idx] + Wc[idx];
    __syncthreads();
  }
}

// ------------------------------------------------------------------
// y = (y_intra + y_inter + D*x) * silu(z), in place in ybuf
// ------------------------------------------------------------------
__global__ __launch_bounds__(256) void combine_kernel(const float* __restrict__ proj,
                                                      const float* __restrict__ Dv,
                                                      float* __restrict__ ybuf) {
  const size_t idx = (size_t)blockIdx.x * 256 + threadIdx.x;
  if (idx >= (size_t)BATCH * SEQLEN * DINNER) return;
  const int p = idx & 63;
  const int h = (idx >> 6) & 31;
  const size_t row = idx >> 11;  // b*SEQLEN + l
  const float z = proj[row * DPROJ + OFF_Z + h * HD + p];
  const float x = proj[row * DPROJ + OFF_X + h * HD + p];
  float y = ybuf[idx] + Dv[h] * x;
  ybuf[idx] = y * (z * sigmoid_f(z));
}

// ------------------------------------------------------------------
extern "C" void kernel_launch(void* const* d_in, const int* in_sizes, int n_in,
                              void* d_out, int out_size, void* d_ws, size_t ws_size,
                              hipStream_t stream) {
  const float* u       = (const float*)d_in[0];
  const float* W_in    = (const float*)d_in[1];
  const float* W_out   = (const float*)d_in[2];
  const float* dt_bias = (const float*)d_in[3];
  const float* B_bias  = (const float*)d_in[4];
  const float* C_bias  = (const float*)d_in[5];
  const float* Bnw     = (const float*)d_in[6];
  const float* Cnw     = (const float*)d_in[7];
  const float* Dv      = (const float*)d_in[8];
  float* out = (float*)d_out;

  float* ws = (float*)d_ws;
  size_t o = 0;
  float* proj   = ws + o;  o += (size_t)BATCH * SEQLEN * DPROJ;
  float* Kbuf   = ws + o;  o += (size_t)BATCH * SEQLEN * NH * DS;
  float* Qbuf   = ws + o;  o += (size_t)BATCH * SEQLEN * NH * DS;
  float* ybuf   = ws + o;  o += (size_t)BATCH * SEQLEN * NH * HD;
  float* Wsum   = ws + o;  o += (size_t)BATCH * NCHUNK * NH * HD * DS;
  float* dtg    = ws + o;  o += (size_t)BATCH * SEQLEN * NH;
  float* g1g    = ws + o;  o += (size_t)BATCH * SEQLEN * NH;
  float* g2g    = ws + o;  o += (size_t)BATCH * SEQLEN * NH;
  float* ellg   = ws + o;  o += (size_t)BATCH * SEQLEN * NH;
  float* cdk    = ws + o;  o += (size_t)BATCH * NCHUNK * NH;
  float* thsum  = ws + o;  o += (size_t)BATCH * NCHUNK * NH * NANG;
  float* thbase = ws + o;  o += (size_t)BATCH * NCHUNK * NH * NANG;
  (void)ws_size; (void)in_sizes; (void)n_in; (void)out_size;

  const int M = BATCH * SEQLEN;  // 4096

  // 1. proj = u @ W_in   (4096 x 1024 x 4480)
  wmma_gemm_f32<<<dim3(DPROJ / 128, M / 128), 256, 0, stream>>>(
      u, W_in, proj, M, DPROJ, DMODEL);
  // 2. RMSNorm of Bg/Cg in place
  rmsnorm_kernel<<<M, 128, 0, stream>>>(Bnw, Cnw, proj);
  // 3. dt/g1/g2/ell
  prep_kernel<<<(BATCH * NCHUNK * NH + 255) / 256, 256, 0, stream>>>(
      proj, dt_bias, dtg, g1g, g2g, ellg);
  // 4. theta cumsum (3 parallel passes) + RoPE
  theta_partial_kernel<<<(BATCH * NCHUNK * NH * NANG + 255) / 256, 256, 0, stream>>>(
      proj, dtg, thsum);
  theta_base_kernel<<<(BATCH * NH * NANG + 255) / 256, 256, 0, stream>>>(thsum, thbase);
  rope_kernel<<<BATCH * NCHUNK * NH, 32, 0, stream>>>(
      proj, dtg, thbase, B_bias, C_bias, Kbuf, Qbuf);
  // 5. per-chunk WMMA (TDM-staged K tile): T, y_intra, Wsum, chunk_decay
  chunk_kernel<<<BATCH * NCHUNK * NH, 256, 0, stream>>>(
      proj, Kbuf, Qbuf, g1g, g2g, ellg, ybuf, Wsum, cdk);
  // 6. inter-chunk scan + y_inter
  scan_kernel<<<BATCH * NH, 256, 0, stream>>>(Qbuf, ellg, Wsum, cdk, ybuf);
  // 7. combine with D*x and silu(z)
  combine_kernel<<<(BATCH * SEQLEN * DINNER) / 256, 256, 0, stream>>>(proj, Dv, ybuf);
  // 8. out = y @ W_out  (4096 x 2048 x 1024)
  wmma_gemm_f32<<<dim3(DMODEL / 128, M / 128), 256, 0, stream>>>(
      ybuf, W_out, out, M, DMODEL, DINNER);
}